// CriticGNN_59047210385711
// MI455X (gfx1250) — compile-verified
//
#include <hip/hip_runtime.h>
#include <hip/hip_bf16.h>

typedef float v2f __attribute__((ext_vector_type(2)));
typedef float v8f __attribute__((ext_vector_type(8)));

// D = A(16x4 f32) * B(4x16 f32) + C(16x16 f32), wave32 WMMA
#define WMMA_F32X4(A, B, C) \
  __builtin_amdgcn_wmma_f32_16x16x4_f32(false, (A), false, (B), (short)0, (C), false, false)

// CDNA5 async global->LDS copy (ASYNCcnt-tracked) + waits. Inline asm is
// portable across ROCm-7.2 / amdgpu-toolchain per the bridge doc. The LDS
// address escapes via ptrtoint and both asms clobber "memory", so the
// compiler must assume the staged buffer is written by the asm.
__device__ __forceinline__ void async_g2l_b128(unsigned lds_off, const void* gaddr) {
  asm volatile("global_load_async_to_lds_b128 %0, %1, off"
               :: "v"(lds_off), "v"((unsigned long long)(uintptr_t)gaddr)
               : "memory");
}
template <int NN>
__device__ __forceinline__ void wait_asynccnt() {
  asm volatile("s_wait_asynccnt %0" :: "i"(NN) : "memory");
}
__device__ __forceinline__ void wait_dscnt0() {
  asm volatile("s_wait_dscnt 0x0" ::: "memory");
}

// ---------------------------------------------------------------------------
// Layer-1 GEMM: for a 16-node tile, compute  hs = x@Ws + bias  and  hn = x@Wn
// x: [N,128], Ws/Wn: [128,16] row-major, hs/hn: [N,16].
// A tile is staged through LDS with double-buffered async b128 copies:
// global side is fully coalesced (8 lanes x 16B per row segment); LDS row
// stride 36 floats makes the WMMA A-reads bank-conflict-free ((m*36+k) mod 64
// -> unique 4-bank slot per row). Per-wave buffers, no block barriers needed
// (ASYNCcnt/DScnt are per-wave; async completion is in-order).
// ---------------------------------------------------------------------------
__global__ __launch_bounds__(256) void gemm_l1(
    const float* __restrict__ x, const float* __restrict__ ws,
    const float* __restrict__ wn, const float* __restrict__ bias,
    float* __restrict__ hs, float* __restrict__ hn, int N)
{
  __shared__ __align__(16) float xs[8][2][16][36];   // 36.9 KB per block
  const int lane = threadIdx.x & 31;
  const int wid  = threadIdx.x >> 5;
  const int ntiles = (N + 15) >> 4;
  const int tile = blockIdx.x * 8 + wid;       // uniform per wave -> EXEC all 1s
  if (tile >= ntiles) return;                  // wave-uniform exit, no block sync
  const int half = lane >> 4;
  const int m = lane & 15;                     // A row / C column index

  // staging pattern: per pass, 32 lanes cover 4 rows x 32 floats (b128 each)
  const int lrow = lane >> 3;                  // 0..3
  const int lcol = (lane & 7) * 4;             // float offset 0..28

  auto stage = [&](int kc, int buf) {
#pragma unroll
    for (int pass = 0; pass < 4; ++pass) {
      const int r = pass * 4 + lrow;
      int grow = tile * 16 + r; if (grow >= N) grow = N - 1;
      const float* gp = x + (long long)grow * 128 + kc * 32 + lcol;
      async_g2l_b128((unsigned)(uintptr_t)&xs[wid][buf][r][lcol], gp);
    }
  };

  const float bv = bias[m];
  v8f cs = {bv, bv, bv, bv, bv, bv, bv, bv};   // bias folded into C
  v8f cn = {0.f, 0.f, 0.f, 0.f, 0.f, 0.f, 0.f, 0.f};

  stage(0, 0);                                 // prologue: chunk 0 -> buf 0
#pragma unroll
  for (int kc = 0; kc < 4; ++kc) {             // K chunks of 32
    const int buf = kc & 1;
    if (kc < 3) {
      wait_dscnt0();                           // buffer reuse: prior reads done
      stage(kc + 1, buf ^ 1);                  // prefetch next chunk
      wait_asynccnt<4>();                      // in-order: current chunk landed
    } else {
      wait_asynccnt<0>();
    }
#pragma unroll
    for (int ks = 0; ks < 8; ++ks) {
      const int k0 = ks * 4 + 2 * half;        // chunk-local K
      const int gk = kc * 32 + k0;             // global K for weights
      v2f a;  a.x  = xs[wid][buf][m][k0];  a.y  = xs[wid][buf][m][k0 + 1];
      v2f bs; bs.x = ws[gk * 16 + m];      bs.y = ws[(gk + 1) * 16 + m];
      v2f bn; bn.x = wn[gk * 16 + m];      bn.y = wn[(gk + 1) * 16 + m];
      cs = WMMA_F32X4(a, bs, cs);
      cn = WMMA_F32X4(a, bn, cn);
    }
  }
#pragma unroll
  for (int j = 0; j < 8; ++j) {
    const int row = tile * 16 + j + 8 * half;
    if (row < N) {
      hs[(long long)row * 16 + m] = cs[j];
      hn[(long long)row * 16 + m] = cn[j];
    }
  }
}

// ---------------------------------------------------------------------------
// Edge aggregation, 16 features: acc[dst] += t[src].  4 threads per edge.
// ---------------------------------------------------------------------------
__global__ __launch_bounds__(256) void edge_agg16(
    const float* __restrict__ t, const int* __restrict__ edges,
    float* __restrict__ acc, int E)
{
  const int idx = blockIdx.x * 256 + threadIdx.x;
  if (idx >= E * 4) return;
  const int e = idx >> 2, q = (idx & 3) * 4;
  const int s = edges[e], d = edges[E + e];
  const float* sp = t + (long long)s * 16 + q;
  float* dp = acc + (long long)d * 16 + q;
  atomicAdd(dp + 0, sp[0]); atomicAdd(dp + 1, sp[1]);
  atomicAdd(dp + 2, sp[2]); atomicAdd(dp + 3, sp[3]);
}

// ---------------------------------------------------------------------------
// Layer-2 GEMM: mol[:, colOff+0..49] = relu(h1)@Ws + bias ; tmp2 = relu(h1)@Wn
// h1: [N,16], Ws/Wn: [16,50] row-major. One wave per (node-tile, col-tile).
// ---------------------------------------------------------------------------
__global__ __launch_bounds__(256) void gemm_l2(
    const float* __restrict__ h1, const float* __restrict__ ws,
    const float* __restrict__ wn, const float* __restrict__ bias,
    float* __restrict__ mol, float* __restrict__ tmp2, int N, int colOff)
{
  const int lane = threadIdx.x & 31;
  const int wid  = threadIdx.x >> 5;
  const int ntiles = (N + 15) >> 4;
  const long long gw = (long long)blockIdx.x * 8 + wid;
  const int tile = (int)(gw >> 2);
  const int ct   = (int)(gw & 3);
  if (tile >= ntiles) return;
  const int half = lane >> 4;
  const int m = lane & 15;
  int nrow = tile * 16 + m; if (nrow >= N) nrow = N - 1;
  const float* hr = h1 + (long long)nrow * 16;
  const int col = ct * 16 + m;
  const bool cv = col < 50;
  const float bv = cv ? bias[col] : 0.f;
  v8f cs = {bv, bv, bv, bv, bv, bv, bv, bv};
  v8f cn = {0.f, 0.f, 0.f, 0.f, 0.f, 0.f, 0.f, 0.f};
#pragma unroll
  for (int ks = 0; ks < 4; ++ks) {
    const int k0 = ks * 4 + 2 * half;
    v2f a;  a.x = fmaxf(hr[k0], 0.f);  a.y = fmaxf(hr[k0 + 1], 0.f);  // fused ReLU
    v2f bs, bn;
    bs.x = cv ? ws[k0 * 50 + col] : 0.f; bs.y = cv ? ws[(k0 + 1) * 50 + col] : 0.f;
    bn.x = cv ? wn[k0 * 50 + col] : 0.f; bn.y = cv ? wn[(k0 + 1) * 50 + col] : 0.f;
    cs = WMMA_F32X4(a, bs, cs);
    cn = WMMA_F32X4(a, bn, cn);
  }
  if (cv) {
#pragma unroll
    for (int j = 0; j < 8; ++j) {
      const int row = tile * 16 + j + 8 * half;
      if (row < N) {
        mol[(long long)row * 100 + colOff + col] = cs[j];
        tmp2[(long long)row * 50 + col]          = cn[j];
      }
    }
  }
}

// ---------------------------------------------------------------------------
// Edge aggregation, 50 features into strided mol: 32 threads per edge.
// ---------------------------------------------------------------------------
__global__ __launch_bounds__(256) void edge_agg50(
    const float* __restrict__ t, const int* __restrict__ edges,
    float* __restrict__ mol, int E, int colOff)
{
  const long long idx = (long long)blockIdx.x * 256 + threadIdx.x;
  if (idx >= (long long)E * 32) return;
  const int e = (int)(idx >> 5);
  const int f = (int)(idx & 31);
  const int s = edges[e], d = edges[E + e];
  const float* sp = t + (long long)s * 50;
  float* dp = mol + (long long)d * 100 + colOff;
  atomicAdd(dp + f, sp[f]);
  if (f + 32 < 50) atomicAdd(dp + f + 32, sp[f + 32]);
}

// ---------------------------------------------------------------------------
// Head: fp = relu(mol@w_in + b_in) [WMMA, staged to LDS], then per-node
// cat(fp, action)@w_hid -> relu -> @w_out + b_out (scalar, tiny FLOPs).
// ---------------------------------------------------------------------------
__global__ __launch_bounds__(256) void head_k(
    const float* __restrict__ mol, const float* __restrict__ action,
    const float* __restrict__ w_in, const float* __restrict__ b_in,
    const float* __restrict__ w_hid, const float* __restrict__ b_hid,
    const float* __restrict__ w_out, const float* __restrict__ b_out,
    float* __restrict__ out, int N)
{
  __shared__ float fpL[8][16][64];             // per-wave 16x60 fp tile (padded)
  const int lane = threadIdx.x & 31;
  const int wid  = threadIdx.x >> 5;
  const int ntiles = (N + 15) >> 4;
  int tile = blockIdx.x * 8 + wid;
  const bool active = tile < ntiles;           // uniform per wave; no early exit
  if (!active) tile = 0;                       // redundant compute, store skipped
  const int half = lane >> 4;
  const int m = lane & 15;
  int nrow = tile * 16 + m; if (nrow >= N) nrow = N - 1;
  const float* mr = mol + (long long)nrow * 100;

  v8f acc[4];
#pragma unroll
  for (int ct = 0; ct < 4; ++ct) {
    const int col = ct * 16 + m;
    const float b = (col < 60) ? b_in[col] : 0.f;
    v8f init = {b, b, b, b, b, b, b, b};
    acc[ct] = init;
  }
  for (int ks = 0; ks < 25; ++ks) {            // K = 100
    const int k0 = ks * 4 + 2 * half;
    v2f a; a.x = mr[k0]; a.y = mr[k0 + 1];
#pragma unroll
    for (int ct = 0; ct < 4; ++ct) {
      const int col = ct * 16 + m;
      const bool cv = col < 60;
      v2f b;
      b.x = cv ? w_in[k0 * 60 + col] : 0.f;
      b.y = cv ? w_in[(k0 + 1) * 60 + col] : 0.f;
      acc[ct] = WMMA_F32X4(a, b, acc[ct]);
    }
  }
#pragma unroll
  for (int ct = 0; ct < 4; ++ct) {
    const int col = ct * 16 + m;
    if (col < 60) {
#pragma unroll
      for (int j = 0; j < 8; ++j)
        fpL[wid][j + 8 * half][col] = fmaxf(acc[ct][j], 0.f);
    }
  }
  __syncthreads();

  if (lane < 16) {                             // one node per lane
    const int node = tile * 16 + lane;
    const bool nv = active && (node < N);
    const int nd = (node < N) ? node : 0;
    float a2[10];
#pragma unroll
    for (int t = 0; t < 10; ++t) a2[t] = b_hid[t];
    for (int k = 0; k < 70; ++k) {
      const float v = (k < 60) ? fpL[wid][lane][k]
                               : action[(long long)nd * 10 + (k - 60)];
#pragma unroll
      for (int t = 0; t < 10; ++t) a2[t] += v * w_hid[k * 10 + t];
    }
    float r = b_out[0];
#pragma unroll
    for (int t = 0; t < 10; ++t) r += fmaxf(a2[t], 0.f) * w_out[t];
    if (nv) out[nd] = r;
  }
}

// ---------------------------------------------------------------------------
extern "C" void kernel_launch(void* const* d_in, const int* in_sizes, int n_in,
                              void* d_out, int out_size, void* d_ws, size_t ws_size,
                              hipStream_t stream)
{
  const float* px   = (const float*)d_in[0];
  const int*   pe   = (const int*)d_in[1];
  const float* lx   = (const float*)d_in[2];
  const int*   le   = (const int*)d_in[3];
  const float* act  = (const float*)d_in[4];
  const float* wp1s = (const float*)d_in[5];
  const float* wp1n = (const float*)d_in[6];
  const float* bp1  = (const float*)d_in[7];
  const float* wp2s = (const float*)d_in[8];
  const float* wp2n = (const float*)d_in[9];
  const float* bp2  = (const float*)d_in[10];
  const float* wl1s = (const float*)d_in[11];
  const float* wl1n = (const float*)d_in[12];
  const float* bl1  = (const float*)d_in[13];
  const float* wl2s = (const float*)d_in[14];
  const float* wl2n = (const float*)d_in[15];
  const float* bl2  = (const float*)d_in[16];
  const float* w_in = (const float*)d_in[17];
  const float* b_in = (const float*)d_in[18];
  const float* w_hid= (const float*)d_in[19];
  const float* b_hid= (const float*)d_in[20];
  const float* w_out= (const float*)d_in[21];
  const float* b_out= (const float*)d_in[22];
  float* out = (float*)d_out;

  const int N = in_sizes[0] / 128;
  const int E = in_sizes[1] / 2;

  float* ws   = (float*)d_ws;
  float* mol  = ws;                          // N*100
  float* h1   = mol  + (size_t)N * 100;      // N*16
  float* tmp1 = h1   + (size_t)N * 16;       // N*16
  float* tmp2 = tmp1 + (size_t)N * 16;       // N*50

  const int ntiles = (N + 15) / 16;
  const dim3 blk(256);
  const int g1 = (ntiles + 7) / 8;
  const int g2 = (ntiles * 4 + 7) / 8;
  const int ge16 = (int)(((long long)E * 4  + 255) / 256);
  const int ge50 = (int)(((long long)E * 32 + 255) / 256);

  // protein branch
  gemm_l1  <<<g1,   blk, 0, stream>>>(px, wp1s, wp1n, bp1, h1, tmp1, N);
  edge_agg16<<<ge16, blk, 0, stream>>>(tmp1, pe, h1, E);
  gemm_l2  <<<g2,   blk, 0, stream>>>(h1, wp2s, wp2n, bp2, mol, tmp2, N, 0);
  edge_agg50<<<ge50, blk, 0, stream>>>(tmp2, pe, mol, E, 0);
  // ligand branch (reuses h1/tmp1/tmp2 — stream-ordered)
  gemm_l1  <<<g1,   blk, 0, stream>>>(lx, wl1s, wl1n, bl1, h1, tmp1, N);
  edge_agg16<<<ge16, blk, 0, stream>>>(tmp1, le, h1, E);
  gemm_l2  <<<g2,   blk, 0, stream>>>(h1, wl2s, wl2n, bl2, mol, tmp2, N, 50);
  edge_agg50<<<ge50, blk, 0, stream>>>(tmp2, le, mol, E, 50);
  // head
  head_k   <<<g1,   blk, 0, stream>>>(mol, act, w_in, b_in, w_hid, b_hid,
                                      w_out, b_out, out, N);
}